// HighOrderAggregator_60301340836383
// MI455X (gfx1250) — compile-verified
//
#include <hip/hip_runtime.h>

#define DIN  128
#define DOUT 128
#define BN_EPS 1e-5f
#define EDGES_PER_WAVE 32
#define TSTRIDE 132   // LDS row stride in floats: bank rotation 4, float2-aligned

typedef float v2f __attribute__((ext_vector_type(2)));
typedef float v8f __attribute__((ext_vector_type(8)));

// ---------------------------------------------------------------------------
// Kernel 1: SpMM  hop1[r,:] += val * feat[c,:]   (rows sorted ascending)
// hop1 lives interleaved in d_out at columns [128,256) of each 256-wide row.
// One wave handles EDGES_PER_WAVE consecutive edges; lane owns 4 features.
// Register-accumulate runs of equal rows, atomic-flush on row change.
// ---------------------------------------------------------------------------
__global__ __launch_bounds__(256) void spmm_kernel(
    const float* __restrict__ feat, const float* __restrict__ vals,
    const int* __restrict__ rows, const int* __restrict__ cols,
    float* __restrict__ out, int E) {
  const int lane = threadIdx.x & 31;
  const int wave = blockIdx.x * (blockDim.x >> 5) + (threadIdx.x >> 5);
  int e0 = wave * EDGES_PER_WAVE;
  if (e0 >= E) return;
  int e1 = e0 + EDGES_PER_WAVE;
  if (e1 > E) e1 = E;

  float4 acc = make_float4(0.f, 0.f, 0.f, 0.f);
  int cur = rows[e0];
  for (int e = e0; e < e1; ++e) {
    const int r = rows[e];
    if (r != cur) {
      float* dst = out + (size_t)cur * 256 + 128 + lane * 4;
      atomicAdd(dst + 0, acc.x); atomicAdd(dst + 1, acc.y);
      atomicAdd(dst + 2, acc.z); atomicAdd(dst + 3, acc.w);
      acc = make_float4(0.f, 0.f, 0.f, 0.f);
      cur = r;
    }
    const float v = vals[e];
    const float4 f = ((const float4*)(feat + (size_t)cols[e] * DIN))[lane];
    acc.x = fmaf(v, f.x, acc.x);
    acc.y = fmaf(v, f.y, acc.y);
    acc.z = fmaf(v, f.z, acc.z);
    acc.w = fmaf(v, f.w, acc.w);
  }
  float* dst = out + (size_t)cur * 256 + 128 + lane * 4;
  atomicAdd(dst + 0, acc.x); atomicAdd(dst + 1, acc.y);
  atomicAdd(dst + 2, acc.z); atomicAdd(dst + 3, acc.w);
}

// ---------------------------------------------------------------------------
// Kernel 2: fused dual GEMM (p0 = relu(feat@W0)+b0, p1 = relu(hop1@W1)+b1)
// via V_WMMA_F32_16X16X4_F32, plus per-column sum / sum-of-squares for BN.
// 512 threads = 16 waves: waves 0-7 -> hop0 (W0), waves 8-15 -> hop1 (W1).
// B operands (the wave's 128x16 slice of W) are hoisted into 64 VGPRs once;
// A tiles are software-pipelined: prefetch tile t+gridDim to registers while
// the WMMA chain for tile t runs out of LDS.
// ---------------------------------------------------------------------------
__global__ __launch_bounds__(512) void gemm_bn_kernel(
    const float* __restrict__ feat, const float* __restrict__ W0,
    const float* __restrict__ W1, const float* __restrict__ b0,
    const float* __restrict__ b1, float* __restrict__ out,
    float* __restrict__ stats, int n_tiles) {
  __shared__ float tA[2][16 * TSTRIDE];

  const int tid  = threadIdx.x;
  const int lane = tid & 31;
  const int wave = tid >> 5;
  const int hop  = wave >> 3;        // 0: feat@W0, 1: hop1@W1
  const int ntl  = wave & 7;         // column tile within hop
  const int m    = lane & 15;        // WMMA row / column-in-tile index
  const int kh   = lane >> 4;        // half-wave (K offset selector)
  const int wcol = ntl * 16 + m;     // column 0..127 within this hop

  const float* __restrict__ W = hop ? W1 : W0;
  const float bv = (hop ? b1 : b0)[wcol];

  const int srow = tid >> 5;         // staging: row 0..15
  const int sk4  = tid & 31;         // staging: float4 index 0..31

  // ---- hoist the wave's B slice of W into registers (loop-invariant) ----
  // B layout for 16x16x4: VGPR j of lane(m,kh) holds W[4*ks + 2*kh + j][wcol]
  v2f breg[32];
#pragma unroll
  for (int ks = 0; ks < 32; ++ks) {
    const int k0 = ks * 4 + kh * 2;
    breg[ks].x = W[(size_t)k0 * DOUT + wcol];
    breg[ks].y = W[(size_t)(k0 + 1) * DOUT + wcol];
  }

  float sum = 0.f, sumsq = 0.f;

  // ---- prologue: load first tile into registers ----
  float4 g0 = make_float4(0.f, 0.f, 0.f, 0.f);
  float4 g1 = make_float4(0.f, 0.f, 0.f, 0.f);
  if (blockIdx.x < n_tiles) {
    const size_t rb = (size_t)blockIdx.x * 16;
    g0 = *(const float4*)(feat + (rb + srow) * DIN + sk4 * 4);
    g1 = *(const float4*)(out + (rb + srow) * 256 + 128 + sk4 * 4);
  }

  for (int t = blockIdx.x; t < n_tiles; t += gridDim.x) {
    const size_t rbase = (size_t)t * 16;

    // --- commit staged registers to LDS ---
    {
      float* d0 = &tA[0][srow * TSTRIDE + sk4 * 4];
      *(float2*)(d0 + 0) = make_float2(g0.x, g0.y);
      *(float2*)(d0 + 2) = make_float2(g0.z, g0.w);
      float* d1 = &tA[1][srow * TSTRIDE + sk4 * 4];
      *(float2*)(d1 + 0) = make_float2(g1.x, g1.y);
      *(float2*)(d1 + 2) = make_float2(g1.z, g1.w);
    }
    __syncthreads();

    // --- prefetch next owned tile (block-uniform condition) ---
    const int tn = t + gridDim.x;
    if (tn < n_tiles) {
      const size_t rb = (size_t)tn * 16;
      g0 = *(const float4*)(feat + (rb + srow) * DIN + sk4 * 4);
      g1 = *(const float4*)(out + (rb + srow) * 256 + 128 + sk4 * 4);
    }

    // --- K loop: 32 x V_WMMA_F32_16X16X4_F32, A from LDS, B from VGPRs ---
    v8f acc = {};
    const float* As = &tA[hop][m * TSTRIDE];
#pragma unroll
    for (int ks = 0; ks < 32; ++ks) {
      const int k0 = ks * 4 + kh * 2;
      const float2 a2 = *(const float2*)(As + k0);  // A[m][k0], A[m][k0+1]
      v2f a; a.x = a2.x; a.y = a2.y;
      acc = __builtin_amdgcn_wmma_f32_16x16x4_f32(
          /*neg_a=*/false, a, /*neg_b=*/false, breg[ks],
          /*c_mod=*/(short)0, acc, /*reuse_a=*/false, /*reuse_b=*/false);
    }

    // --- relu + bias, store D tile, accumulate BN stats ---
    float s = 0.f, q = 0.f;
#pragma unroll
    for (int i = 0; i < 8; ++i) {
      const float v = fmaxf(acc[i], 0.f) + bv;
      const size_t row = rbase + i + kh * 8;      // C/D layout: M = i + 8*(lane>=16)
      out[row * 256 + hop * 128 + wcol] = v;
      s += v;
      q = fmaf(v, v, q);
    }
    sum += s;
    sumsq += q;
    __syncthreads();   // protect LDS before next staging
  }

  // lanes L and L^16 hold the same output column -> fold, then one atomic
  sum   += __shfl_xor(sum,   16, 32);
  sumsq += __shfl_xor(sumsq, 16, 32);
  if (kh == 0) {
    atomicAdd(&stats[hop * 128 + wcol],       sum);
    atomicAdd(&stats[256 + hop * 128 + wcol], sumsq);
  }
}

// ---------------------------------------------------------------------------
// Kernel 3: BatchNorm finalize: out = (out - mean)*rsqrt(var+eps)*gamma + beta
// ---------------------------------------------------------------------------
__global__ __launch_bounds__(256) void bn_kernel(
    const float* __restrict__ stats, const float* __restrict__ gamma,
    const float* __restrict__ beta, float* __restrict__ out,
    int total4, float invN) {
  __shared__ float sc[256], sh[256];
  {
    const int c = threadIdx.x;        // blockDim.x == 256
    const float mean = stats[c] * invN;
    const float var  = fmaf(-mean, mean, stats[256 + c] * invN);
    const float s    = gamma[c] * rsqrtf(var + BN_EPS);
    sc[c] = s;
    sh[c] = fmaf(-mean, s, beta[c]);
  }
  __syncthreads();

  float4* o4 = (float4*)out;
  const int stride = gridDim.x * blockDim.x;
  for (int i = blockIdx.x * blockDim.x + threadIdx.x; i < total4; i += stride) {
    const int c4 = (i & 63) << 2;     // 64 float4 per 256-wide row
    float4 v = o4[i];
    v.x = fmaf(v.x, sc[c4 + 0], sh[c4 + 0]);
    v.y = fmaf(v.y, sc[c4 + 1], sh[c4 + 1]);
    v.z = fmaf(v.z, sc[c4 + 2], sh[c4 + 2]);
    v.w = fmaf(v.w, sc[c4 + 3], sh[c4 + 3]);
    o4[i] = v;
  }
}

// ---------------------------------------------------------------------------
extern "C" void kernel_launch(void* const* d_in, const int* in_sizes, int n_in,
                              void* d_out, int out_size, void* d_ws, size_t ws_size,
                              hipStream_t stream) {
  const float* feat  = (const float*)d_in[0];
  const float* vals  = (const float*)d_in[1];
  const float* W0    = (const float*)d_in[2];
  const float* W1    = (const float*)d_in[3];
  const float* b0    = (const float*)d_in[4];
  const float* b1    = (const float*)d_in[5];
  const float* gamma = (const float*)d_in[6];
  const float* beta  = (const float*)d_in[7];
  const int*   rows  = (const int*)d_in[8];
  const int*   cols  = (const int*)d_in[9];

  const int E  = in_sizes[1];
  const int Nn = in_sizes[0] / DIN;

  float* out   = (float*)d_out;
  float* stats = (float*)d_ws;      // 512 floats: [0,256) sums, [256,512) sumsq

  // zero hop1 accumulation region (entire out buffer) + BN stat accumulators
  hipMemsetAsync(d_out, 0, (size_t)out_size * sizeof(float), stream);
  hipMemsetAsync(d_ws, 0, 512 * sizeof(float), stream);

  // 1) SpMM into out[:,128:256]
  const int waves = (E + EDGES_PER_WAVE - 1) / EDGES_PER_WAVE;
  const int spmm_blocks = (waves + 7) / 8;
  spmm_kernel<<<spmm_blocks, 256, 0, stream>>>(feat, vals, rows, cols, out, E);

  // 2) fused dual GEMM + BN statistics
  const int n_tiles = Nn / 16;      // N = 100000 -> 6250 tiles
  const int gblocks = n_tiles < 1024 ? n_tiles : 1024;
  gemm_bn_kernel<<<gblocks, 512, 0, stream>>>(feat, W0, W1, b0, b1, out, stats, n_tiles);

  // 3) BN normalize in place
  bn_kernel<<<1024, 256, 0, stream>>>(stats, gamma, beta, out, Nn * 64,
                                      1.0f / (float)Nn);
}